// MyLinearLayer_20890720928459
// MI455X (gfx1250) — compile-verified
//
#include <hip/hip_runtime.h>
#include <hip/hip_bf16.h>

typedef float v2f __attribute__((ext_vector_type(2)));
typedef float v8f __attribute__((ext_vector_type(8)));

#define HW 49
#define CPB 64              // channels per block (2 waves, 1 channel per thread for I/O)
#define PAIRS (CPB / 2)     // 32 channel-pairs per block
#define PSTRIDE 100         // floats per pair region: 2*49 data + 2 pad (zero / dump)

// Offset inside a pair's 100-float LDS region for element (r,c) of the
// 16x16 block-diagonal matrix: block0 = rows/cols 0..6 (channel 0),
// block1 = rows/cols 7..13 (channel 1), everything else -> zero/dump pad at 98.
__device__ __forceinline__ int off_blockdiag(int r, int c) {
  if (r < 7 && c < 7) return r * 7 + c;
  if (r >= 7 && r < 14 && c >= 7 && c < 14) return 49 + (r - 7) * 7 + (c - 7);
  return 98;
}

__global__ __launch_bounds__(CPB) void batched7x7_wmma_f32(
    const float* __restrict__ x, const float* __restrict__ w,
    const float* __restrict__ bias, float* __restrict__ out, int C) {
  __shared__ float Xs[PAIRS * PSTRIDE];  // per-channel 7x7 X, already transposed
  __shared__ float Ws[PAIRS * PSTRIDE];  // per-channel 7x7 W
  __shared__ float Os[PAIRS * PSTRIDE];  // per-channel 7x7 result

  const int tid = threadIdx.x;
  const int cblk = blockIdx.x * CPB;

  // Zero the pads (read as the "0" entries of the block-diagonal operands).
  for (int p = tid; p < PAIRS; p += CPB) {
    Xs[p * PSTRIDE + 98] = 0.f;
    Xs[p * PSTRIDE + 99] = 0.f;
    Ws[p * PSTRIDE + 98] = 0.f;
    Ws[p * PSTRIDE + 99] = 0.f;
  }

  // Stage x: contiguous global reads, transposed 7x7 per channel into LDS.
  // Xc[c,i,j] = x[c, j, i]  ->  Xs[pair][sub*49 + i*7 + j]
  {
    const float* xb = x + (size_t)cblk * HW;
    for (int idx = tid; idx < CPB * HW; idx += CPB) {
      int c = idx / HW;
      int r = idx - c * HW;    // r = h*7 + w
      int h = r / 7;
      int wc = r - h * 7;
      Xs[(c >> 1) * PSTRIDE + (c & 1) * HW + (wc * 7 + h)] = xb[idx];
    }
  }

  // Stage weight: lane = channel -> perfectly coalesced strided reads.
  // Wc[c,j,k] = weight[(j*7+k)*C + c] -> Ws[pair][sub*49 + j*7 + k]
  {
    const int c = cblk + tid;
    const int base = (tid >> 1) * PSTRIDE + (tid & 1) * HW;
    for (int p = 0; p < HW; ++p)
      Ws[base + p] = w[(size_t)p * C + c];
  }

  __syncthreads();

  // Per-lane, loop-invariant gather/scatter offsets for the WMMA operands.
  // A 16x4 f32 layout: lanes 0-15 M=0..15; VGPR r holds K = 2*(lane>=16) + r.
  // B 4x16 f32 layout: VGPR r holds row K = 2*(lane>=16) + r, N = lane%16.
  // C/D 16x16 f32:     VGPR v holds M = v (lanes 0-15) / v+8 (lanes 16-31), N = lane%16.
  const int lane = tid & 31;
  const int lm = lane & 15;
  const int hi = lane >> 4;
  int aOff[4][2], bOff[4][2], dOff[8];
#pragma unroll
  for (int g = 0; g < 4; ++g) {
#pragma unroll
    for (int r = 0; r < 2; ++r) {
      int k = 4 * g + 2 * hi + r;
      aOff[g][r] = off_blockdiag(lm, k);  // A[m=lm][k]
      bOff[g][r] = off_blockdiag(k, lm);  // B[k][n=lm]
    }
  }
#pragma unroll
  for (int v = 0; v < 8; ++v) {
    int M = (lane < 16) ? v : v + 8;
    dOff[v] = off_blockdiag(M, lm);       // D[M][N=lm] -> out position, or dump
  }

  // Each wave handles 16 channel-pairs; 4 chained f32 WMMAs cover K = 0..15
  // (K=14,15 read the zero pad).
  const int waveId = tid >> 5;
#pragma unroll 1
  for (int p = 0; p < PAIRS / 2; ++p) {
    const int pr = waveId * (PAIRS / 2) + p;
    const float* xp = &Xs[pr * PSTRIDE];
    const float* wp = &Ws[pr * PSTRIDE];
    float* op = &Os[pr * PSTRIDE];

    v8f acc = {0.f, 0.f, 0.f, 0.f, 0.f, 0.f, 0.f, 0.f};
#pragma unroll
    for (int g = 0; g < 4; ++g) {
      v2f a, b;
      a.x = xp[aOff[g][0]];
      a.y = xp[aOff[g][1]];
      b.x = wp[bOff[g][0]];
      b.y = wp[bOff[g][1]];
      acc = __builtin_amdgcn_wmma_f32_16x16x4_f32(
          /*neg_a=*/false, a, /*neg_b=*/false, b,
          /*c_mod=*/(short)0, acc, /*reuse_a=*/false, /*reuse_b=*/false);
    }
#pragma unroll
    for (int v = 0; v < 8; ++v)
      op[dOff[v]] = acc[v];
  }

  __syncthreads();

  // Writeout: lane = channel -> coalesced bias read + out store.
  {
    const int c = cblk + tid;
    const int base = (tid >> 1) * PSTRIDE + (tid & 1) * HW;
    for (int p = 0; p < HW; ++p) {
      size_t gi = (size_t)p * C + c;
      out[gi] = Os[base + p] + bias[gi];
    }
  }
}

extern "C" void kernel_launch(void* const* d_in, const int* in_sizes, int n_in,
                              void* d_out, int out_size, void* d_ws, size_t ws_size,
                              hipStream_t stream) {
  const float* x = (const float*)d_in[0];     // (1, C, 7, 7)
  const float* w = (const float*)d_in[1];     // (49, C)
  const float* b = (const float*)d_in[2];     // (49, C)
  float* out = (float*)d_out;                 // (49, C)
  const int C = in_sizes[0] / HW;

  const int grid = C / CPB;
  batched7x7_wmma_f32<<<grid, CPB, 0, stream>>>(x, w, b, out, C);
}